// MovementEmbeddingModule_29368986370686
// MI455X (gfx1250) — compile-verified
//
#include <hip/hip_runtime.h>

typedef float v2f __attribute__((ext_vector_type(2)));
typedef float v8f __attribute__((ext_vector_type(8)));

#define BS    4
#define DNUM  8
#define KNUM  10
#define K1    11
#define CH_OUT 66      // K1 * (1 + 2 + 3)
#define H     128
#define W     128
#define HW    (H * W)
#define INV2VAR 50.0f  // 0.5 / KP_VARIANCE(=0.01)

__device__ __forceinline__ float gauss1d(int i, float kp) {
    // grid coord: 2*i/(n-1) - 1
    float g = __fmaf_rn((float)i, 2.0f / 127.0f, -1.0f);
    float dlt = g - kp;
    return __expf(-INV2VAR * dlt * dlt);
}

// ---------------------------------------------------------------------------
// Kernel 1: separable normalization sums. S[p] = Sx * Sy  (p = (b,dd,k), 320)
// ---------------------------------------------------------------------------
__global__ void me_sums_kernel(const float* __restrict__ kpd,
                               float* __restrict__ S) {
    int p  = blockIdx.x;                 // 0..319
    int kk = p % KNUM;
    int dd = (p / KNUM) % DNUM;
    int b  = p / (KNUM * DNUM);
    const float* kp = kpd + (((b * DNUM + dd) * KNUM + kk) << 1);
    float kx = kp[0], ky = kp[1];

    int t = threadIdx.x;                 // 0..127
    __shared__ float sx[128];
    __shared__ float sy[128];
    sx[t] = gauss1d(t, kx);
    sy[t] = gauss1d(t, ky);
    __syncthreads();
    #pragma unroll
    for (int s = 64; s > 0; s >>= 1) {
        if (t < s) { sx[t] += sx[t + s]; sy[t] += sy[t + s]; }
        __syncthreads();
    }
    if (t == 0) S[p] = sx[0] * sy[0];
}

// ---------------------------------------------------------------------------
// Kernel 2: heat planes (k=1..10, j=0) via WMMA rank-1 outer product.
// One wave32 computes one 16x16 tile: D = (ey*invS) (x) ex  using
// V_WMMA_F32_16X16X4_F32 with only K=0 column of A / row of B populated.
// ---------------------------------------------------------------------------
__global__ void me_heat_kernel(const float* __restrict__ kpd,
                               const float* __restrict__ S,
                               float* __restrict__ out) {
    int wid  = blockIdx.x * (blockDim.x >> 5) + (threadIdx.x >> 5);
    int lane = threadIdx.x & 31;
    int tile = wid & 63;                 // 8x8 tiles of 16x16 per plane
    int p    = wid >> 6;                 // 0..319
    int kk   = p % KNUM;
    int dd   = (p / KNUM) % DNUM;
    int b    = p / (KNUM * DNUM);
    int tx   = (tile & 7) << 4;
    int ty   = (tile >> 3) << 4;

    const float* kp = kpd + (((b * DNUM + dd) * KNUM + kk) << 1);
    float kx   = kp[0], ky = kp[1];
    float invS = 1.0f / S[p];

    int  l15 = lane & 15;
    bool lo  = lane < 16;

    // A (16x4 f32): lanes 0-15 VGPR0 hold (M=lane, K=0); everything else 0.
    // B (4x16 f32): lanes 0-15 VGPR0 hold (K=0, N=lane); everything else 0.
    float ey = gauss1d(ty + l15, ky) * invS;
    float ex = gauss1d(tx + l15, kx);
    v2f A; A[0] = lo ? ey : 0.0f; A[1] = 0.0f;
    v2f B; B[0] = lo ? ex : 0.0f; B[1] = 0.0f;
    v8f C = {};
    v8f D = __builtin_amdgcn_wmma_f32_16x16x4_f32(
        /*neg_a=*/false, A, /*neg_b=*/false, B,
        /*c_mod=*/(short)0, C, /*reuse_a=*/false, /*reuse_b=*/false);

    // D layout: lane n holds rows m0..m0+7 at column n%16
    int ch = (kk + 1) * 6;               // j = 0
    float* dst = out + ((size_t)(b * CH_OUT + ch) * DNUM + dd) * HW;
    int m0 = lo ? 0 : 8;
    #pragma unroll
    for (int v = 0; v < 8; ++v)
        dst[(ty + m0 + v) * W + tx + l15] = D[v];
}

// ---------------------------------------------------------------------------
// Kernel 3: constant planes. Per (b,dd): k=0 -> j=0,1,2 zeros; k>=1 -> j=1,2
// = kp_source - kp_driving. 736 planes, float4 stores.
// ---------------------------------------------------------------------------
__global__ void me_const_kernel(const float* __restrict__ kpd,
                                const float* __restrict__ kps,
                                float* __restrict__ out) {
    int bi    = blockIdx.x;
    int plane = bi >> 4;                 // 0..735
    int seg   = bi & 15;
    int q     = plane % 23;
    int bd    = plane / 23;
    int b = bd >> 3, dd = bd & 7;

    int k, j; float val;
    if (q < 3) { k = 0; j = q; val = 0.0f; }
    else {
        int q2 = q - 3;
        k = (q2 >> 1) + 1;
        j = (q2 & 1) + 1;
        int idx = (((b * DNUM + dd) * KNUM + (k - 1)) << 1) + (j - 1);
        val = kps[idx] - kpd[idx];
    }
    int ch = k * 6 + j;
    float4* dst = (float4*)(out + ((size_t)(b * CH_OUT + ch) * DNUM + dd) * HW);
    dst[seg * 256 + threadIdx.x] = make_float4(val, val, val, val);
}

// ---------------------------------------------------------------------------
// Kernel 4: deformed source. Constant per-plane shift => fixed 2x2-tap blend
// with integer offsets and constant weights; zero padding outside.
// ---------------------------------------------------------------------------
__global__ void me_deform_kernel(const float* __restrict__ src,
                                 const float* __restrict__ kpd,
                                 const float* __restrict__ kps,
                                 float* __restrict__ out) {
    int bi    = blockIdx.x;
    int plane = bi >> 6;                 // 0..1055
    int seg   = bi & 63;                 // 2 rows per block
    int q     = plane % 33;
    int bd    = plane / 33;
    int b = bd >> 3, dd = bd & 7;
    int k = q / 3, ch = q % 3;

    float dx = 0.0f, dy = 0.0f;
    if (k > 0) {
        int idx = ((b * DNUM + dd) * KNUM + (k - 1)) << 1;
        dx = kps[idx]     - kpd[idx];
        dy = kps[idx + 1] - kpd[idx + 1];
    }
    float sx = dx * 63.5f, sy = dy * 63.5f;      // *(W-1)/2
    float fox = floorf(sx), foy = floorf(sy);
    int   ox = (int)fox,    oy = (int)foy;
    float fx = sx - fox,    fy = sy - foy;
    float wx0 = 1.0f - fx,  wy0 = 1.0f - fy;

    int x = threadIdx.x & 127;
    int y = (seg << 1) + (threadIdx.x >> 7);
    const float* sp = src + (size_t)(b * 3 + ch) * HW;

    int x0i = x + ox, x1i = x0i + 1;
    int y0i = y + oy, y1i = y0i + 1;
    float mx0 = ((unsigned)x0i < (unsigned)W) ? 1.0f : 0.0f;
    float mx1 = ((unsigned)x1i < (unsigned)W) ? 1.0f : 0.0f;
    float my0 = ((unsigned)y0i < (unsigned)H) ? 1.0f : 0.0f;
    float my1 = ((unsigned)y1i < (unsigned)H) ? 1.0f : 0.0f;
    int cx0 = min(max(x0i, 0), W - 1), cx1 = min(max(x1i, 0), W - 1);
    int cy0 = min(max(y0i, 0), H - 1), cy1 = min(max(y1i, 0), H - 1);

    __builtin_prefetch(sp + cy1 * W + cx0, 0, 1);  // global_prefetch_b8

    float v00 = sp[cy0 * W + cx0] * (mx0 * my0);
    float v01 = sp[cy0 * W + cx1] * (mx1 * my0);
    float v10 = sp[cy1 * W + cx0] * (mx0 * my1);
    float v11 = sp[cy1 * W + cx1] * (mx1 * my1);

    float r = (v00 * wx0 + v01 * fx) * wy0 + (v10 * wx0 + v11 * fx) * fy;

    int cho = k * 6 + 3 + ch;
    out[((size_t)(b * CH_OUT + cho) * DNUM + dd) * HW + y * W + x] = r;
}

// ---------------------------------------------------------------------------
extern "C" void kernel_launch(void* const* d_in, const int* in_sizes, int n_in,
                              void* d_out, int out_size, void* d_ws, size_t ws_size,
                              hipStream_t stream) {
    const float* src = (const float*)d_in[0];   // (4,3,1,128,128)
    const float* kpd = (const float*)d_in[1];   // (4,8,10,2)
    const float* kps = (const float*)d_in[2];   // (4,8,10,2)
    float* out = (float*)d_out;                 // (4,66,8,128,128)
    float* S   = (float*)d_ws;                  // 320 floats

    me_sums_kernel<<<BS * DNUM * KNUM, 128, 0, stream>>>(kpd, S);
    me_heat_kernel<<<(BS * DNUM * KNUM * 64) / 8, 256, 0, stream>>>(kpd, S, out);
    me_const_kernel<<<BS * DNUM * 23 * 16, 256, 0, stream>>>(kpd, kps, out);
    me_deform_kernel<<<BS * DNUM * 33 * 64, 256, 0, stream>>>(src, kpd, kps, out);
}